// DeepGATConvClassifier2_91027536871851
// MI455X (gfx1250) — compile-verified
//
#include <hip/hip_runtime.h>
#include <hip/hip_bf16.h>

#define NN   50000
#define EE   800000
#define TE   (EE + NN)      // edges + self loops
#define IND  128
#define HEADS 4
#define HID  64
#define C1   256            // HEADS*HID
#define OUTD 64

typedef float v2f __attribute__((ext_vector_type(2)));
typedef float v8f __attribute__((ext_vector_type(8)));

// ---------------------------------------------------------------------------
// WMMA f32 GEMM: C[M,Nc] = A[M,K] * B[K,Nc] (+bias) (+relu). One wave per
// 16x16 tile, K stepped by 4 using V_WMMA_F32_16X16X4_F32.
// Requires M%16==0, K%4==0, Nc%16==0 (true for all calls here).
// ---------------------------------------------------------------------------
__global__ void wmma_gemm_f32(const float* __restrict__ A,
                              const float* __restrict__ B,
                              float* __restrict__ C,
                              const float* __restrict__ bias,
                              int M, int K, int Nc, int act) {
  int wave  = (blockIdx.x * blockDim.x + threadIdx.x) >> 5;
  int lane  = threadIdx.x & 31;
  int ntile = Nc >> 4;
  int tm = wave / ntile;
  int tn = wave - tm * ntile;
  if (tm * 16 >= M) return;               // whole-wave uniform exit
  int row0 = tm << 4, col0 = tn << 4;
  int mlane = lane & 15;
  int khalf = (lane >> 4) << 1;           // 0 or 2

  v8f acc = {};
  const float* arow = A + (size_t)(row0 + mlane) * K;
  for (int k = 0; k < K; k += 4) {
    v2f a, b;
    const float* ap = arow + k + khalf;
    a.x = ap[0];
    a.y = ap[1];
    const float* bp = B + (size_t)(k + khalf) * Nc + col0 + mlane;
    b.x = bp[0];
    b.y = bp[Nc];
    acc = __builtin_amdgcn_wmma_f32_16x16x4_f32(false, a, false, b,
                                                (short)0, acc, false, false);
  }
  int rbase = row0 + ((lane >> 4) << 3);  // lanes 16-31 hold M = j+8
  int col   = col0 + mlane;
  float bv = bias ? bias[col] : 0.f;
  #pragma unroll
  for (int j = 0; j < 8; ++j) {
    float v = acc[j] + bv;
    if (act == 1) v = fmaxf(v, 0.f);
    C[(size_t)(rbase + j) * Nc + col] = v;
  }
}

// ---------------------------------------------------------------------------
// BatchNorm (train-mode, biased variance) + fused activation
// ---------------------------------------------------------------------------
__global__ void bn_stats(const float* __restrict__ X, int M, int Nc,
                         float* __restrict__ mean, float* __restrict__ var) {
  int col = blockIdx.x;
  __shared__ float ssum[256], ssq[256];
  float s = 0.f, q = 0.f;
  for (int r = threadIdx.x; r < M; r += blockDim.x) {
    float v = X[(size_t)r * Nc + col];
    s += v; q += v * v;
  }
  ssum[threadIdx.x] = s; ssq[threadIdx.x] = q;
  __syncthreads();
  for (int off = blockDim.x >> 1; off > 0; off >>= 1) {
    if ((int)threadIdx.x < off) {
      ssum[threadIdx.x] += ssum[threadIdx.x + off];
      ssq[threadIdx.x]  += ssq[threadIdx.x + off];
    }
    __syncthreads();
  }
  if (threadIdx.x == 0) {
    float mu = ssum[0] / (float)M;
    mean[col] = mu;
    var[col]  = ssq[0] / (float)M - mu * mu;
  }
}

// act: 0 none, 1 relu, 2 elu (alpha=1)
__global__ void bn_apply(float* __restrict__ X, long long total, int Nc,
                         const float* __restrict__ mean,
                         const float* __restrict__ var,
                         const float* __restrict__ g,
                         const float* __restrict__ b, int act) {
  long long i = (long long)blockIdx.x * blockDim.x + threadIdx.x;
  if (i >= total) return;
  int col = (int)(i % Nc);
  float v = (X[i] - mean[col]) * rsqrtf(var[col] + 1e-5f) * g[col] + b[col];
  if (act == 1)      v = fmaxf(v, 0.f);
  else if (act == 2) v = v > 0.f ? v : (expf(v) - 1.f);
  X[i] = v;
}

__global__ void fill_bias_rows(float* __restrict__ out,
                               const float* __restrict__ bias,
                               long long total, int Nc) {
  long long i = (long long)blockIdx.x * blockDim.x + threadIdx.x;
  if (i >= total) return;
  out[i] = bias[i % Nc];
}

// ---------------------------------------------------------------------------
// GAT attention pieces
// ---------------------------------------------------------------------------
__global__ void gat_scores(const float* __restrict__ Hm,
                           const float* __restrict__ asrc,
                           const float* __restrict__ adst,
                           float* __restrict__ es, float* __restrict__ ed,
                           int n, int heads, int dim) {
  int i = blockIdx.x * blockDim.x + threadIdx.x;
  if (i >= n * heads) return;
  int node = i / heads, h_ = i - node * heads;
  const float* hp = Hm + ((size_t)node * heads + h_) * dim;
  const float* as = asrc + h_ * dim;
  const float* ad = adst + h_ * dim;
  float s = 0.f, d = 0.f;
  for (int k = 0; k < dim; ++k) { float v = hp[k]; s += v * as[k]; d += v * ad[k]; }
  es[i] = s; ed[i] = d;
}

__device__ __forceinline__ unsigned f32_ord(float f) {
  unsigned u = __float_as_uint(f);
  return (u & 0x80000000u) ? ~u : (u | 0x80000000u);
}
__device__ __forceinline__ float ord_f32(unsigned u) {
  return __uint_as_float((u & 0x80000000u) ? (u & 0x7FFFFFFFu) : ~u);
}

__global__ void edge_max(const int* __restrict__ src, const int* __restrict__ dst,
                         const float* __restrict__ es, const float* __restrict__ ed,
                         unsigned* __restrict__ emax, int heads) {
  long long i = (long long)blockIdx.x * blockDim.x + threadIdx.x;
  long long tot = (long long)TE * heads;
  if (i >= tot) return;
  int e  = (int)(i / heads);
  int h_ = (int)(i - (long long)e * heads);
  int s = e < EE ? src[e] : e - EE;
  int d = e < EE ? dst[e] : e - EE;
  float v = es[s * heads + h_] + ed[d * heads + h_];
  v = v > 0.f ? v : 0.2f * v;                       // LeakyReLU(0.2)
  atomicMax(emax + d * heads + h_, f32_ord(v));
}

__global__ void edge_expsum(const int* __restrict__ src, const int* __restrict__ dst,
                            const float* __restrict__ es, const float* __restrict__ ed,
                            const unsigned* __restrict__ emax,
                            float* __restrict__ wbuf, float* __restrict__ denom,
                            int heads) {
  long long i = (long long)blockIdx.x * blockDim.x + threadIdx.x;
  long long tot = (long long)TE * heads;
  if (i >= tot) return;
  int e  = (int)(i / heads);
  int h_ = (int)(i - (long long)e * heads);
  int s = e < EE ? src[e] : e - EE;
  int d = e < EE ? dst[e] : e - EE;
  float v = es[s * heads + h_] + ed[d * heads + h_];
  v = v > 0.f ? v : 0.2f * v;
  float w = expf(v - ord_f32(emax[d * heads + h_]));
  wbuf[i] = w;
  atomicAdd(denom + d * heads + h_, w);
}

__global__ void edge_aggregate(const int* __restrict__ src, const int* __restrict__ dst,
                               const float* __restrict__ Hm,
                               const float* __restrict__ wbuf,
                               const float* __restrict__ denom,
                               float* __restrict__ agg, int heads, int dim) {
  long long i = (long long)blockIdx.x * blockDim.x + threadIdx.x;
  long long tot = (long long)TE * heads * dim;
  if (i >= tot) return;
  int f = (int)(i % dim);
  long long eh = i / dim;
  int h_ = (int)(eh % heads);
  int e  = (int)(eh / heads);
  int s = e < EE ? src[e] : e - EE;
  int d = e < EE ? dst[e] : e - EE;
  float alpha = wbuf[eh] / (denom[d * heads + h_] + 1e-16f);
  atomicAdd(agg + ((size_t)d * heads + h_) * dim + f,
            Hm[((size_t)s * heads + h_) * dim + f] * alpha);
}

// ---------------------------------------------------------------------------
// GCN pieces
// ---------------------------------------------------------------------------
__global__ void gcn_deg(const int* __restrict__ dst, float* __restrict__ deg) {
  int e = blockIdx.x * blockDim.x + threadIdx.x;
  if (e >= TE) return;
  int d = e < EE ? dst[e] : e - EE;
  atomicAdd(deg + d, 1.0f);
}

__global__ void gcn_dinv(float* __restrict__ deg) {
  int i = blockIdx.x * blockDim.x + threadIdx.x;
  if (i >= NN) return;
  deg[i] = rsqrtf(fmaxf(deg[i], 1e-12f));
}

__global__ void gcn_aggregate(const int* __restrict__ src, const int* __restrict__ dst,
                              const float* __restrict__ H2,
                              const float* __restrict__ dinv,
                              float* __restrict__ agg, int dim) {
  long long i = (long long)blockIdx.x * blockDim.x + threadIdx.x;
  long long tot = (long long)TE * dim;
  if (i >= tot) return;
  int f = (int)(i % dim);
  int e = (int)(i / dim);
  int s = e < EE ? src[e] : e - EE;
  int d = e < EE ? dst[e] : e - EE;
  float norm = dinv[s] * dinv[d];
  atomicAdd(agg + (size_t)d * dim + f, H2[(size_t)s * dim + f] * norm);
}

// ---------------------------------------------------------------------------
// Host-side launch
// ---------------------------------------------------------------------------
static inline void launch_gemm(const float* A, const float* B, float* C,
                               const float* bias, int M, int K, int Nc, int act,
                               hipStream_t stream) {
  long long waves = (long long)(M / 16) * (Nc / 16);
  int block = 128;                           // 4 waves / block
  long long grid = (waves * 32 + block - 1) / block;
  wmma_gemm_f32<<<(int)grid, block, 0, stream>>>(A, B, C, bias, M, K, Nc, act);
}

extern "C" void kernel_launch(void* const* d_in, const int* in_sizes, int n_in,
                              void* d_out, int out_size, void* d_ws, size_t ws_size,
                              hipStream_t stream) {
  const float* x       = (const float*)d_in[0];
  const int*   ei      = (const int*)  d_in[1];
  const int*   srcI    = ei;
  const int*   dstI    = ei + EE;
  const float* W1      = (const float*)d_in[3];
  const float* asrc1   = (const float*)d_in[4];
  const float* adst1   = (const float*)d_in[5];
  const float* b1      = (const float*)d_in[6];
  const float* g1      = (const float*)d_in[7];
  const float* beta1   = (const float*)d_in[8];
  const float* Wg      = (const float*)d_in[9];
  const float* bg      = (const float*)d_in[10];
  const float* g2      = (const float*)d_in[11];
  const float* beta2   = (const float*)d_in[12];
  const float* W2      = (const float*)d_in[13];
  const float* asrc2   = (const float*)d_in[14];
  const float* adst2   = (const float*)d_in[15];
  const float* b2      = (const float*)d_in[16];
  const float* g3      = (const float*)d_in[17];
  const float* beta3   = (const float*)d_in[18];
  const float* Wp1     = (const float*)d_in[19];
  const float* bp1     = (const float*)d_in[20];
  const float* Wp2     = (const float*)d_in[21];
  const float* bp2     = (const float*)d_in[22];
  float* out = (float*)d_out;

  // ---- workspace layout (floats) ----
  float* ws = (float*)d_ws;
  size_t off = 0;
  float*    h1    = ws + off; off += (size_t)NN * C1;   // 12.8M  x@W1
  float*    agg1  = ws + off; off += (size_t)NN * C1;   // 12.8M  GAT1 out / later z1
  float*    ebuf  = ws + off; off += (size_t)TE * HEADS;// 3.4M   edge weights / later agg2
  float*    es1   = ws + off; off += (size_t)NN * HEADS;
  float*    ed1   = ws + off; off += (size_t)NN * HEADS;
  unsigned* emax1 = (unsigned*)(ws + off); off += (size_t)NN * HEADS;
  float*    denom = ws + off; off += (size_t)NN * HEADS;
  float*    deg   = ws + off; off += (size_t)NN;
  float*    meanb = ws + off; off += 256;
  float*    varb  = ws + off; off += 256;

  // region reuse
  float* h2   = h1;                    // [NN,64]  GCN gemm
  float* agg2 = ebuf;                  // [NN,64]  GCN aggregation (TE*4 >= NN*64)
  float* h3   = h1 + (size_t)NN * HID; // [NN,64]  GAT2 gemm
  float* wb2  = agg1;                  // [TE]     GAT2 edge weights
  float* agg3 = h1 + 2 * (size_t)NN * HID; // [NN,64] GAT2 out
  float* z1   = agg1;                  // [NN,128] projector hidden

  const int B = 256;
  auto blocks = [](long long n, int b) { return (int)((n + b - 1) / b); };

  // ================= GAT layer 1 =================
  launch_gemm(x, W1, h1, nullptr, NN, IND, C1, 0, stream);
  gat_scores<<<blocks((long long)NN * HEADS, B), B, 0, stream>>>(
      h1, asrc1, adst1, es1, ed1, NN, HEADS, HID);
  hipMemsetAsync(emax1, 0, (size_t)NN * HEADS * 4, stream);
  hipMemsetAsync(denom, 0, (size_t)NN * HEADS * 4, stream);
  fill_bias_rows<<<blocks((long long)NN * C1, B), B, 0, stream>>>(
      agg1, b1, (long long)NN * C1, C1);
  edge_max<<<blocks((long long)TE * HEADS, B), B, 0, stream>>>(
      srcI, dstI, es1, ed1, emax1, HEADS);
  edge_expsum<<<blocks((long long)TE * HEADS, B), B, 0, stream>>>(
      srcI, dstI, es1, ed1, emax1, ebuf, denom, HEADS);
  edge_aggregate<<<blocks((long long)TE * HEADS * HID, B), B, 0, stream>>>(
      srcI, dstI, h1, ebuf, denom, agg1, HEADS, HID);
  bn_stats<<<C1, 256, 0, stream>>>(agg1, NN, C1, meanb, varb);
  bn_apply<<<blocks((long long)NN * C1, B), B, 0, stream>>>(
      agg1, (long long)NN * C1, C1, meanb, varb, g1, beta1, 2 /*elu*/);

  // ================= GCN layer =================
  launch_gemm(agg1, Wg, h2, nullptr, NN, C1, HID, 0, stream);
  hipMemsetAsync(deg, 0, (size_t)NN * 4, stream);
  gcn_deg<<<blocks((long long)TE, B), B, 0, stream>>>(dstI, deg);
  gcn_dinv<<<blocks((long long)NN, B), B, 0, stream>>>(deg);
  fill_bias_rows<<<blocks((long long)NN * HID, B), B, 0, stream>>>(
      agg2, bg, (long long)NN * HID, HID);
  gcn_aggregate<<<blocks((long long)TE * HID, B), B, 0, stream>>>(
      srcI, dstI, h2, deg, agg2, HID);
  bn_stats<<<HID, 256, 0, stream>>>(agg2, NN, HID, meanb, varb);
  bn_apply<<<blocks((long long)NN * HID, B), B, 0, stream>>>(
      agg2, (long long)NN * HID, HID, meanb, varb, g2, beta2, 1 /*relu*/);

  // ================= GAT layer 2 (heads=1) =================
  launch_gemm(agg2, W2, h3, nullptr, NN, HID, OUTD, 0, stream);
  gat_scores<<<blocks((long long)NN, B), B, 0, stream>>>(
      h3, asrc2, adst2, es1, ed1, NN, 1, OUTD);
  hipMemsetAsync(emax1, 0, (size_t)NN * 4, stream);
  hipMemsetAsync(denom, 0, (size_t)NN * 4, stream);
  fill_bias_rows<<<blocks((long long)NN * OUTD, B), B, 0, stream>>>(
      agg3, b2, (long long)NN * OUTD, OUTD);
  edge_max<<<blocks((long long)TE, B), B, 0, stream>>>(
      srcI, dstI, es1, ed1, emax1, 1);
  edge_expsum<<<blocks((long long)TE, B), B, 0, stream>>>(
      srcI, dstI, es1, ed1, emax1, wb2, denom, 1);
  edge_aggregate<<<blocks((long long)TE * OUTD, B), B, 0, stream>>>(
      srcI, dstI, h3, wb2, denom, agg3, 1, OUTD);
  bn_stats<<<OUTD, 256, 0, stream>>>(agg3, NN, OUTD, meanb, varb);
  bn_apply<<<blocks((long long)NN * OUTD, B), B, 0, stream>>>(
      agg3, (long long)NN * OUTD, OUTD, meanb, varb, g3, beta3, 2 /*elu*/);

  // ================= projector =================
  launch_gemm(agg3, Wp1, z1, bp1, NN, OUTD, 128, 1 /*relu*/, stream);
  launch_gemm(z1, Wp2, out, bp2, NN, 128, OUTD, 0, stream);

  (void)in_sizes; (void)n_in; (void)out_size; (void)ws_size;
}